// NUFFT_47536698032130
// MI455X (gfx1250) — compile-verified
//
#include <hip/hip_runtime.h>
#include <stdint.h>

// ---------------------------------------------------------------------------
// Type-2 NUDFT on gfx1250 (MI455X), fused two-stage separable contraction.
//
// Stage 1 (complex GEMM per (b,c) slice, on V_WMMA_F32_16X16X4_F32):
//   tmp[k,x] = sum_y exp(-i*om_y[k]*(y-64)) * image[x,y]
// Stage 2 (fused per-lane epilogue + shfl_xor butterfly reduction):
//   out[k]   = sum_x tmp[k,x] * exp(-i*om_x[k]*(x-64))
//
// v2 changes vs round 1:
//  * LDS deinterleaved into re/im planes, layout [x][y] pitch 132:
//      - one ds_load_b64 per lane IS the full 4x16 WMMA B-fragment
//        (lanes 0-15 -> K={0,1}, lanes 16-31 -> K={2,3}), no v_mov repack
//      - pitch 132: even (b64 align) and 4-bank stride -> conflict-free
//  * staging via global_load_async_to_lds_b128 + s_wait_asynccnt (no VGPR
//    round trip; exercises the gfx1250 ASYNCcnt path)
// ---------------------------------------------------------------------------

typedef __attribute__((ext_vector_type(2))) float v2f;
typedef __attribute__((ext_vector_type(8))) float v8f;

#define DIM_H   128
#define DIM_W   128
#define K_TOT   2048
#define N_BC    16              // B*C
#define K_PER_WG 128            // 8 waves * 16 k-rows
#define PITCH   132             // floats per x-row in LDS
#define PLANE   (DIM_H * PITCH) // 16896 floats per plane

// D = A(16x4 f32) * B(4x16 f32) + C(16x16 f32)
__device__ __forceinline__ v8f wmma_f32(v2f a, v2f b, v8f c) {
    return __builtin_amdgcn_wmma_f32_16x16x4_f32(
        false, a, false, b, (short)0, c, false, false);
}

__global__ __launch_bounds__(256, 1)
void nufft_t2_wmma_kernel(const float* __restrict__ img_re,
                          const float* __restrict__ img_im,
                          const float* __restrict__ traj,
                          float* __restrict__ out /* interleaved complex64 */) {
    extern __shared__ float lds[];
    float* sRe = lds;           // [x][y] pitch PITCH
    float* sIm = lds + PLANE;

    const int tid  = threadIdx.x;
    const int lane = tid & 31;
    const int wave = tid >> 5;
    const int bc    = blockIdx.x >> 4;     // 16 k-tiles per slice
    const int ktile = blockIdx.x & 15;
    const int k0w   = ktile * K_PER_WG + wave * 16;

    const float* slice_re = img_re + (size_t)bc * DIM_H * DIM_W;
    const float* slice_im = img_im + (size_t)bc * DIM_H * DIM_W;
    const float* om_x = traj + (size_t)bc * 2 * K_TOT;   // traj[b,c,0,:]
    const float* om_y = om_x + K_TOT;                    // traj[b,c,1,:]

    // ---- Async-stage the slice into LDS planes (no transpose needed) -------
    // 16384 floats/plane, 256 threads * 16B -> 16 iterations, 2 planes.
    // Per-lane: 16B from global (coalesced) -> 16B to LDS at pitch-132 row.
    for (int i = 0; i < (DIM_H * DIM_W) / (256 * 4); ++i) {
        const int f = (i * 256 + tid) * 4;     // first float of this lane's quad
        const int x = f >> 7;                  // f / W
        const int y = f & (DIM_W - 1);         // f % W  (quad stays in-row)
        const unsigned ldsRe = (unsigned)(uintptr_t)&sRe[x * PITCH + y];
        const unsigned ldsIm = (unsigned)(uintptr_t)&sIm[x * PITCH + y];
        const unsigned goff  = (unsigned)(f * 4);   // byte offset from slice base
        asm volatile("global_load_async_to_lds_b128 %0, %1, %2"
                     :: "v"(ldsRe), "v"(goff), "s"(slice_re) : "memory");
        asm volatile("global_load_async_to_lds_b128 %0, %1, %2"
                     :: "v"(ldsIm), "v"(goff), "s"(slice_im) : "memory");
    }
    asm volatile("s_wait_asynccnt 0" ::: "memory");
    __syncthreads();

    const int m  = lane & 15;   // A-row (k) / B-col (x) sub-index
    const int hi = lane >> 4;   // upper half-wave: K+2 / M+8 per WMMA layout

    const float omy = om_y[k0w + m];

    v8f zero = {0.f, 0.f, 0.f, 0.f, 0.f, 0.f, 0.f, 0.f};
    v8f accRe[8], accIm[8];     // 8 x-tiles of 16 -> full H=128 per wave
#pragma unroll
    for (int t = 0; t < 8; ++t) { accRe[t] = zero; accIm[t] = zero; }

    // ---- Stage 1: K-loop over y in chunks of 4 (f32 WMMA depth) ------------
    for (int yc = 0; yc < DIM_W; yc += 4) {
        // A-fragment phases in native 16x4 layout:
        //   VGPR0: (M=m, K=2*hi), VGPR1: (M=m, K=2*hi+1)
        const int ya = yc + 2 * hi;
        float c0, s0, c1, s1;
        __sincosf(omy * (float)(ya - 64), &s0, &c0);
        __sincosf(omy * (float)(ya - 63), &s1, &c1);
        const v2f aRe = {c0, c1};      // Re(py)  =  cos
        const v2f aPs = {s0, s1};      // -Im(py) =  sin
        const v2f aNs = {-s0, -s1};    //  Im(py) = -sin

#pragma unroll
        for (int xt = 0; xt < 8; ++xt) {
            const int x = xt * 16 + m;
            // One b64 per lane == the whole 4x16 B-fragment (see header note).
            const v2f bRe = *(const v2f*)(sRe + x * PITCH + ya);
            const v2f bIm = *(const v2f*)(sIm + x * PITCH + ya);
            // tmp_re += cos*Ir + sin*Ii ; tmp_im += cos*Ii - sin*Ir
            accRe[xt] = wmma_f32(aRe, bRe, accRe[xt]);
            accRe[xt] = wmma_f32(aPs, bIm, accRe[xt]);
            accIm[xt] = wmma_f32(aRe, bIm, accIm[xt]);
            accIm[xt] = wmma_f32(aNs, bRe, accIm[xt]);
        }
    }

    // ---- Stage 2 (fused): apply px and reduce over x -----------------------
    // C/D layout: VGPR v, lane -> element (k = k0w + v + 8*hi, x = 16*xt + m)
    float2* out2 = (float2*)out;
#pragma unroll
    for (int v = 0; v < 8; ++v) {
        const int k = k0w + 8 * hi + v;
        const float omx = om_x[k];
        float pr = 0.f, pi = 0.f;
#pragma unroll
        for (int xt = 0; xt < 8; ++xt) {
            const int x = xt * 16 + m;
            float c, s;
            __sincosf(omx * (float)(x - 64), &s, &c);
            const float tr = accRe[xt][v];
            const float ti = accIm[xt][v];
            pr += tr * c + ti * s;     // Re(tmp * (c - i s))
            pi += ti * c - tr * s;     // Im(tmp * (c - i s))
        }
#pragma unroll
        for (int off = 1; off < 16; off <<= 1) {
            pr += __shfl_xor(pr, off, 32);
            pi += __shfl_xor(pi, off, 32);
        }
        if (m == 0) {
            out2[(size_t)bc * K_TOT + k] = make_float2(pr, pi);
        }
    }
}

extern "C" void kernel_launch(void* const* d_in, const int* in_sizes, int n_in,
                              void* d_out, int out_size, void* d_ws, size_t ws_size,
                              hipStream_t stream) {
    const float* img_re = (const float*)d_in[0];   // (B,C,H,W) f32
    const float* img_im = (const float*)d_in[1];   // (B,C,H,W) f32
    const float* traj   = (const float*)d_in[2];   // (B,C,2,K) f32
    float* out = (float*)d_out;                    // (B,C,K) complex64 interleaved

    const dim3 grid(N_BC * (K_TOT / K_PER_WG));    // 256 workgroups
    const dim3 block(256);                         // 8 waves (wave32)
    const size_t lds_bytes = 2u * PLANE * sizeof(float);  // 135168 B

    nufft_t2_wmma_kernel<<<grid, block, lds_bytes, stream>>>(img_re, img_im, traj, out);
}